// NoseHoverThermostat_90305982365940
// MI455X (gfx1250) — compile-verified
//
#include <hip/hip_runtime.h>
#include <hip/hip_bf16.h>

// Nose-Hoover chain thermostat (massive, per-atom-per-dim chains).
// Elementwise over 12M independent chains -> HBM-bandwidth-bound streaming
// kernel (~624 MB @ 23.3 TB/s ~= 27 us floor). No matrix structure, so no
// WMMA; instead we hit the gfx1250 memory path: coalesced wave32 loads +
// global_prefetch_b8 lookahead into the 192 MB L2.

#define NHC_CHAIN 3
#define NHC_MULTI 2
#define NHC_ORDER 3
#define NHC_KBT   300.0f

struct __align__(4) f3x { float x, y, z; };

__global__ __launch_bounds__(256) void nhc_massive_kernel(
    const float* __restrict__ momenta,        // [total]   (= N*3 flat)
    const float* __restrict__ atom_masses,    // [total]
    const float* __restrict__ velocities,     // [total*CHAIN]
    const float* __restrict__ forces,         // [total*CHAIN]
    const float* __restrict__ thermo_masses,  // [total*CHAIN]
    const float* __restrict__ dof,            // [total]
    const float* __restrict__ time_step,      // [ORDER]
    float*       __restrict__ out,            // [total]
    int total)
{
    const float LOG2E = 1.4426950408889634f;
    // Uniform address + uniform index -> scalar (SMEM) loads into SGPRs.
    const float ts_arr[NHC_ORDER] = { time_step[0], time_step[1], time_step[2] };

    const int stride = (int)(gridDim.x * blockDim.x);
    for (int e = (int)(blockIdx.x * blockDim.x + threadIdx.x); e < total; e += stride) {
        // ---- prefetch next grid-stride tile (global_prefetch_b8) ----
        const int en = e + stride;
        if (en < total) {
            const size_t en3 = (size_t)en * NHC_CHAIN;
            __builtin_prefetch(&momenta[en],        0, 0);
            __builtin_prefetch(&atom_masses[en],    0, 0);
            __builtin_prefetch(&dof[en],            0, 0);
            __builtin_prefetch(&velocities[en3],    0, 0);
            __builtin_prefetch(&forces[en3],        0, 0);
            __builtin_prefetch(&thermo_masses[en3], 0, 0);
        }

        // ---- stream in this element ----
        const size_t e3 = (size_t)e * NHC_CHAIN;
        const float p  = momenta[e];
        const float am = atom_masses[e];
        const float d  = dof[e];
        const f3x vv = *(const f3x*)&velocities[e3];
        const f3x ff = *(const f3x*)&forces[e3];
        const f3x mm = *(const f3x*)&thermo_masses[e3];

        const float ke  = p * p * __builtin_amdgcn_rcpf(am);
        const float dkt = d * NHC_KBT;
        const float rm0 = __builtin_amdgcn_rcpf(mm.x);
        const float rm1 = __builtin_amdgcn_rcpf(mm.y);
        const float rm2 = __builtin_amdgcn_rcpf(mm.z);

        float v0 = vv.x, v1 = vv.y, v2 = vv.z;
        float f1 = ff.y, f2 = ff.z;
        float f0 = (ke - dkt) * rm0;
        float scale = 1.0f;

        // MULTI * ORDER = 6 fully unrolled Yoshida-Suzuki substeps.
        #pragma unroll
        for (int s = 0; s < NHC_MULTI * NHC_ORDER; ++s) {
            const float ts  = ts_arr[s % NHC_ORDER];
            const float qts = 0.25f * ts;
            const float h8  = -0.125f * LOG2E * ts;  // exp(-0.125*ts*x) = exp2(h8*x)
            const float h2  = -0.5f   * LOG2E * ts;  // exp(-0.5*ts*x)   = exp2(h2*x)

            // tail half-kick
            v2 = fmaf(f2, qts, v2);

            // backward sweep: c = 1, then c = 0 (each coeff uses just-updated v[c+1])
            float c1 = __builtin_amdgcn_exp2f(h8 * v2);
            v1 = fmaf(v1 * c1, c1, f1 * c1 * qts);
            float c0 = __builtin_amdgcn_exp2f(h8 * v1);
            v0 = fmaf(v0 * c0, c0, f0 * c0 * qts);

            // momentum-scale accumulation and chain-head force refresh
            scale = scale * __builtin_amdgcn_exp2f(h2 * v0);
            f0 = fmaf(scale * scale, ke, -dkt) * rm0;

            // forward sweep: c = 0 then c = 1, refreshing downstream forces
            c0 = __builtin_amdgcn_exp2f(h8 * v1);
            v0 = fmaf(v0 * c0, c0, f0 * c0 * qts);
            f1 = fmaf(mm.x * v0, v0, -NHC_KBT) * rm1;

            c1 = __builtin_amdgcn_exp2f(h8 * v2);
            v1 = fmaf(v1 * c1, c1, f1 * c1 * qts);
            f2 = fmaf(mm.y * v1, v1, -NHC_KBT) * rm2;

            // tail half-kick
            v2 = fmaf(f2, qts, v2);
        }

        out[e] = p * scale;
    }
}

extern "C" void kernel_launch(void* const* d_in, const int* in_sizes, int n_in,
                              void* d_out, int out_size, void* d_ws, size_t ws_size,
                              hipStream_t stream) {
    (void)n_in; (void)d_ws; (void)ws_size; (void)out_size;
    const float* momenta       = (const float*)d_in[0];
    const float* atom_masses   = (const float*)d_in[1];
    const float* velocities    = (const float*)d_in[2];
    const float* forces        = (const float*)d_in[3];
    const float* thermo_masses = (const float*)d_in[4];
    const float* dof           = (const float*)d_in[5];
    const float* time_step     = (const float*)d_in[6];
    float* out = (float*)d_out;

    const int total = in_sizes[0];  // N*3 flat elements
    const int threads = 256;
    int blocks = (total + threads - 1) / threads;
    if (blocks > 8192) blocks = 8192;  // grid-stride; ~5.7 elems/thread, 8 MB prefetch lookahead
    if (blocks < 1) blocks = 1;

    nhc_massive_kernel<<<blocks, threads, 0, stream>>>(
        momenta, atom_masses, velocities, forces, thermo_masses, dof,
        time_step, out, total);
}